// REFINE_10548439679706
// MI455X (gfx1250) — compile-verified
//
#include <hip/hip_runtime.h>
#include <math.h>

#define N_NODES   1024
#define IN_F      512
#define N_HEADS   4
#define N_HID     32
#define OUT_F     (N_HEADS * N_HID)   // 128
#define NEG_SLOPE 0.2f

typedef float v2f __attribute__((ext_vector_type(2)));
typedef float v8f __attribute__((ext_vector_type(8)));

// ---------------------------------------------------------------------------
// Kernel 1: C[1024x128] = A[1024x512] @ W[512x128] via V_WMMA_F32_16X16X4_F32
// grid (64 row-tiles, 8 col-tiles), block = 32 (one wave per 16x16 tile)
// ---------------------------------------------------------------------------
__global__ void gat_gemm_f32(const float* __restrict__ A,
                             const float* __restrict__ W,
                             float* __restrict__ C) {
    const int lane = threadIdx.x;          // 0..31
    const int m    = lane & 15;            // row (A) / col (B)
    const int hi   = lane >> 4;            // 0: K0/K1, 1: K2/K3
    const int koff = hi * 2;
    const int r0   = blockIdx.x * 16;
    const int c0   = blockIdx.y * 16;

    v8f acc = {};
    const float* arow = A + (r0 + m) * IN_F;

    for (int k = 0; k < IN_F; k += 4) {
        // A fragment: lane holds A[r0+m][k+koff], A[r0+m][k+koff+1]
        v2f a = *(const v2f*)(arow + k + koff);            // 8B aligned
        // B fragment: v0 = W[k+koff][c0+m], v1 = W[k+koff+1][c0+m]
        v2f b;
        b.x = W[(k + koff    ) * OUT_F + c0 + m];
        b.y = W[(k + koff + 1) * OUT_F + c0 + m];
        acc = __builtin_amdgcn_wmma_f32_16x16x4_f32(
            false, a, false, b, (short)0, acc, false, false);
    }

    // C layout: VGPR r -> row (r + 8*hi), col = m
    #pragma unroll
    for (int r = 0; r < 8; ++r) {
        C[(r0 + r + 8 * hi) * OUT_F + c0 + m] = acc[r];
    }
}

// ---------------------------------------------------------------------------
// Kernel 2: fused GATv2 attention for 8 dest rows x 1 head per block.
//   e[i,j] = sum_f aw[f]*lrelu(gl[j,f] + gr[i,f]); mask; softmax over j;
//   h'[i,:] = sum_j a[i,j] * gr[j,:]  (WMMA aggregation); ELU.
// grid (128 row-blocks, 4 heads), block = 256 (8 waves)
// ---------------------------------------------------------------------------
__global__ void gat_attn(const float* __restrict__ gl,
                         const float* __restrict__ gr,
                         const int*   __restrict__ adj,
                         const float* __restrict__ attn_w,
                         float* __restrict__ out) {
    __shared__ float sP[8][N_NODES];    // 32 KB: e scores, then probabilities
    __shared__ float sGr[8][N_HID];     // dest-side g_r rows
    __shared__ float sAw[N_HID];
    __shared__ float sRed[8][8];
    __shared__ float sStat[8];          // row max
    __shared__ float sRecip[8];         // 1/sum

    const int t    = threadIdx.x;
    const int wave = t >> 5;
    const int lane = t & 31;
    const int i0   = blockIdx.x * 8;
    const int head = blockIdx.y;
    const int hofs = head * N_HID;

    // stage g_r[i0..i0+7, head, :] and attn_w
    {
        int r = t >> 5, f = t & 31;
        sGr[r][f] = gr[(i0 + r) * OUT_F + hofs + f];
        if (t < N_HID) sAw[t] = attn_w[t];
    }
    __syncthreads();

    // attn_w in registers (float4 x 8)
    float4 aw4[8];
    #pragma unroll
    for (int q = 0; q < 8; ++q) aw4[q] = *(const float4*)&sAw[q * 4];

    // ---- phase 1: e scores, per-thread row maxima ----
    float rm[8];
    #pragma unroll
    for (int r = 0; r < 8; ++r) rm[r] = -INFINITY;

    for (int c = 0; c < 4; ++c) {
        const int j = t + c * 256;
        const float4* glv = (const float4*)(gl + j * OUT_F + hofs);
        float4 g4[8];
        #pragma unroll
        for (int q = 0; q < 8; ++q) g4[q] = glv[q];

        #pragma unroll
        for (int r = 0; r < 8; ++r) {
            float acc = 0.f;
            const float4* grv = (const float4*)&sGr[r][0];
            #pragma unroll
            for (int q = 0; q < 8; ++q) {
                float4 gv = grv[q];
                float x0 = g4[q].x + gv.x, x1 = g4[q].y + gv.y;
                float x2 = g4[q].z + gv.z, x3 = g4[q].w + gv.w;
                x0 = (x0 >= 0.f) ? x0 : NEG_SLOPE * x0;
                x1 = (x1 >= 0.f) ? x1 : NEG_SLOPE * x1;
                x2 = (x2 >= 0.f) ? x2 : NEG_SLOPE * x2;
                x3 = (x3 >= 0.f) ? x3 : NEG_SLOPE * x3;
                acc = fmaf(aw4[q].x, x0, acc);
                acc = fmaf(aw4[q].y, x1, acc);
                acc = fmaf(aw4[q].z, x2, acc);
                acc = fmaf(aw4[q].w, x3, acc);
            }
            float e = (adj[(i0 + r) * N_NODES + j] != 0) ? acc : -INFINITY;
            sP[r][j] = e;
            rm[r] = fmaxf(rm[r], e);
        }
    }

    // reduce row maxima: wave shuffle, then cross-wave via LDS
    #pragma unroll
    for (int r = 0; r < 8; ++r) {
        float v = rm[r];
        for (int off = 16; off > 0; off >>= 1)
            v = fmaxf(v, __shfl_xor(v, off, 32));
        if (lane == 0) sRed[r][wave] = v;
    }
    __syncthreads();
    if (t < 8) {
        float mx = -INFINITY;
        #pragma unroll
        for (int w = 0; w < 8; ++w) mx = fmaxf(mx, sRed[t][w]);
        sStat[t] = mx;
    }
    __syncthreads();

    // ---- phase 2: p = exp(e - max), row sums ----
    float rs[8];
    #pragma unroll
    for (int r = 0; r < 8; ++r) rs[r] = 0.f;
    for (int c = 0; c < 4; ++c) {
        const int j = t + c * 256;
        #pragma unroll
        for (int r = 0; r < 8; ++r) {
            float p = expf(sP[r][j] - sStat[r]);
            sP[r][j] = p;
            rs[r] += p;
        }
    }
    #pragma unroll
    for (int r = 0; r < 8; ++r) {
        float v = rs[r];
        for (int off = 16; off > 0; off >>= 1)
            v += __shfl_xor(v, off, 32);
        if (lane == 0) sRed[r][wave] = v;
    }
    __syncthreads();
    if (t < 8) {
        float s = 0.f;
        #pragma unroll
        for (int w = 0; w < 8; ++w) s += sRed[t][w];
        sRecip[t] = 1.0f / s;
    }
    __syncthreads();

    // ---- phase 3: aggregation  H[8,32] = P[8,1024] @ gr[1024, head,:] ----
    // waves 0,1 each own a 16-col N tile; A rows 8..15 are zero-padded.
    if (wave < 2) {
        const int n    = lane & 15;
        const int hi   = lane >> 4;
        const int koff = hi * 2;
        const int f0   = hofs + wave * 16;
        const bool rowOk = (n < 8);            // A rows 8..15 -> zero

        v8f acc = {};
        for (int k = 0; k < N_NODES; k += 4) {
            v2f a;
            a.x = rowOk ? sP[n & 7][k + koff    ] : 0.f;
            a.y = rowOk ? sP[n & 7][k + koff + 1] : 0.f;
            v2f b;
            b.x = gr[(k + koff    ) * OUT_F + f0 + n];
            b.y = gr[(k + koff + 1) * OUT_F + f0 + n];
            acc = __builtin_amdgcn_wmma_f32_16x16x4_f32(
                false, a, false, b, (short)0, acc, false, false);
        }

        // C fragment: VGPR r -> row (r + 8*hi); only rows 0..7 are real
        if (hi == 0) {
            #pragma unroll
            for (int r = 0; r < 8; ++r) {
                float v = acc[r] * sRecip[r];
                v = (v > 0.f) ? v : expm1f(v);          // ELU
                out[(i0 + r) * OUT_F + f0 + n] = v;
            }
        }
    }
}

// ---------------------------------------------------------------------------
extern "C" void kernel_launch(void* const* d_in, const int* in_sizes, int n_in,
                              void* d_out, int out_size, void* d_ws, size_t ws_size,
                              hipStream_t stream) {
    const float* h      = (const float*)d_in[0];
    const int*   adj    = (const int*  )d_in[1];
    const float* W_l    = (const float*)d_in[2];
    const float* W_r    = (const float*)d_in[3];
    const float* attn_w = (const float*)d_in[4];
    float*       out    = (float*)d_out;

    float* g_l = (float*)d_ws;                       // 1024*128 floats
    float* g_r = g_l + N_NODES * OUT_F;              // 1024*128 floats

    dim3 ggrid(N_NODES / 16, OUT_F / 16);
    gat_gemm_f32<<<ggrid, 32, 0, stream>>>(h, W_l, g_l);
    gat_gemm_f32<<<ggrid, 32, 0, stream>>>(h, W_r, g_r);

    dim3 agrid(N_NODES / 8, N_HEADS);
    gat_attn<<<agrid, 256, 0, stream>>>(g_l, g_r, adj, attn_w, out);
}